// GAT_69063074119744
// MI455X (gfx1250) — compile-verified
//
#include <hip/hip_runtime.h>
#include <hip/hip_bf16.h>

typedef __attribute__((ext_vector_type(2))) float v2f;
typedef __attribute__((ext_vector_type(8))) float v8f;

#define HID 128   // X_DIM == HIDDEN == 128 (layout baked into wave tiling)

// ---------------------------------------------------------------- prep: deg=1 (self loop), rootnode[i]=rootindex[batch[i]]
__global__ void k_prep(const int* __restrict__ batch, const int* __restrict__ rootidx,
                       float* __restrict__ deg, int* __restrict__ rootn, int N) {
    int i = blockIdx.x * blockDim.x + threadIdx.x;
    if (i < N) { deg[i] = 1.0f; rootn[i] = rootidx[batch[i]]; }
}

// ---------------------------------------------------------------- deg[dst] += 1 per edge
__global__ void k_deg(const int* __restrict__ eidx, float* __restrict__ deg, int E) {
    int e = blockIdx.x * blockDim.x + threadIdx.x;
    if (e < E) unsafeAtomicAdd(&deg[eidx[E + e]], 1.0f);
}

// ---------------------------------------------------------------- dinv = rsqrt(deg) in place (deg >= 1 always)
__global__ void k_dinv(float* __restrict__ deg, int N) {
    int i = blockIdx.x * blockDim.x + threadIdx.x;
    if (i < N) deg[i] = rsqrtf(deg[i]);
}

// ---------------------------------------------------------------- GEMM via V_WMMA_F32_16X16X4_F32
// MODE 0: out[M,128] = A[M,K] @ W[K,128]
// MODE 1: out[M,128] = relu(concat(nr1[M,128], x[rootn[M],128])) @ W[256,128]
// One wave (32 lanes) computes one 16x16 tile. ISA fp32 layouts:
//   A 16x4 : lanes 0-15 M=0..15 {VGPR0=K0,VGPR1=K1}; lanes 16-31 {K2,K3}
//   B 4x16 : lanes 0-15 N=0..15 {VGPR0=K0,VGPR1=K1}; lanes 16-31 {K2,K3}
//   C 16x16: VGPR r: lanes 0-15 -> M=r, lanes 16-31 -> M=r+8
template <int MODE>
__global__ void k_gemm_wmma(const float* __restrict__ A,
                            const float* __restrict__ x, const int* __restrict__ rootn,
                            const float* __restrict__ W,
                            float* __restrict__ out, int M, int K) {
    const int lane = threadIdx.x;          // 0..31
    const int half = lane >> 4;            // 0 | 1
    const int l15  = lane & 15;
    const int m0   = blockIdx.x * 16;
    const int n0   = (blockIdx.y * blockDim.y + threadIdx.y) * 16;  // 0..112
    if (m0 >= M) return;                   // wave-uniform

    const int mrow = m0 + l15;
    const float* arow = (MODE == 0) ? (A + (size_t)mrow * K) : (A + (size_t)mrow * HID);
    const float* xrow = nullptr;
    if (MODE == 1) xrow = x + (size_t)rootn[mrow] * HID;

    v8f acc = {};
    for (int k = 0; k < K; k += 4) {
        const int ka = k + 2 * half;       // this half-wave's K pair
        v2f a, b;
        if (MODE == 0) {
            a.x = arow[ka];
            a.y = arow[ka + 1];
        } else {
            float a0, a1;
            if (ka < HID) { a0 = arow[ka];        a1 = arow[ka + 1]; }
            else          { a0 = xrow[ka - HID];  a1 = xrow[ka - HID + 1]; }
            a.x = fmaxf(a0, 0.0f);
            a.y = fmaxf(a1, 0.0f);
        }
        b.x = W[(size_t)ka * HID + n0 + l15];
        b.y = W[(size_t)(ka + 1) * HID + n0 + l15];
        acc = __builtin_amdgcn_wmma_f32_16x16x4_f32(
            /*neg_a=*/false, a, /*neg_b=*/false, b,
            /*c_mod=*/(short)0, acc, /*reuse_a=*/false, /*reuse_b=*/false);
    }
    float* orow = out + (size_t)(m0 + 8 * half) * HID + n0 + l15;
#pragma unroll
    for (int r = 0; r < 8; ++r) orow[(size_t)r * HID] = acc[r];
}

// ---------------------------------------------------------------- out[i, j] = bias[j] + h[i,j]*dinv[i]^2  (self-loop + bias init)
__global__ void k_selfbias(const float* __restrict__ h, const float* __restrict__ dinv,
                           const float* __restrict__ bias, float* __restrict__ out,
                           int N, int ostride) {
    int t = blockIdx.x * blockDim.x + threadIdx.x;
    if (t >= N * HID) return;
    int i = t >> 7, j = t & (HID - 1);
    float di = dinv[i];
    out[(size_t)i * ostride + j] = bias[j] + h[(size_t)i * HID + j] * di * di;
}

// ---------------------------------------------------------------- edge scatter: out[dst,:] += h[src,:] * dinv[src]*dinv[dst]
// one wave per edge, float4 per lane (32*4 = 128 features)
__global__ void k_scatter(const int* __restrict__ eidx, const float* __restrict__ h,
                          const float* __restrict__ dinv, float* __restrict__ out,
                          int E, int ostride) {
    int t = blockIdx.x * blockDim.x + threadIdx.x;
    int e = t >> 5, lane = t & 31;
    if (e >= E) return;
    int s = eidx[e], d = eidx[E + e];
    float w = dinv[s] * dinv[d];
    const float4 v = *(const float4*)(h + (size_t)s * HID + lane * 4);
    float* o = out + (size_t)d * ostride + lane * 4;
    unsafeAtomicAdd(o + 0, v.x * w);
    unsafeAtomicAdd(o + 1, v.y * w);
    unsafeAtomicAdd(o + 2, v.z * w);
    unsafeAtomicAdd(o + 3, v.w * w);
}

// ---------------------------------------------------------------- Xf[i,0:128] = relu(Xf[i,0:128]); Xf[i,128:256] = nr1[rootn[i],:]
__global__ void k_relu_fill(float* __restrict__ Xf, const float* __restrict__ nr1,
                            const int* __restrict__ rootn, int N) {
    int t = blockIdx.x * blockDim.x + threadIdx.x;
    if (t >= N * HID) return;
    int i = t >> 7, j = t & (HID - 1);
    size_t base = (size_t)i * (2 * HID);
    Xf[base + j] = fmaxf(Xf[base + j], 0.0f);
    Xf[base + HID + j] = nr1[(size_t)rootn[i] * HID + j];
}

__global__ void k_zero(float* __restrict__ p, int n) {
    int i = blockIdx.x * blockDim.x + threadIdx.x;
    if (i < n) p[i] = 0.0f;
}

// ---------------------------------------------------------------- segment-sum pool over batch (+ counts)
__global__ void k_pool(const float* __restrict__ Xf, const int* __restrict__ batch,
                       float* __restrict__ pool, float* __restrict__ cnt, int N) {
    int t = blockIdx.x * blockDim.x + threadIdx.x;
    if (t >= N * 64) return;                 // 64 float4 chunks of 256 features
    int i = t >> 6, c4 = (t & 63) << 2;
    int g = batch[i];
    const float4 v = *(const float4*)(Xf + (size_t)i * (2 * HID) + c4);
    float* p = pool + (size_t)g * (2 * HID) + c4;
    unsafeAtomicAdd(p + 0, v.x);
    unsafeAtomicAdd(p + 1, v.y);
    unsafeAtomicAdd(p + 2, v.z);
    unsafeAtomicAdd(p + 3, v.w);
    if (c4 == 0) unsafeAtomicAdd(&cnt[g], 1.0f);
}

// ---------------------------------------------------------------- out[g,c] = (pool[g,:]/max(cnt,1)) . fcW[:,c] + fcb[c]
__global__ void k_fc(const float* __restrict__ pool, const float* __restrict__ cnt,
                     const float* __restrict__ fcW, const float* __restrict__ fcb,
                     float* __restrict__ out, int NG, int NC) {
    int t = blockIdx.x * blockDim.x + threadIdx.x;
    if (t >= NG * NC) return;
    int g = t / NC, c = t % NC;
    float inv = 1.0f / fmaxf(cnt[g], 1.0f);
    float acc = fcb[c];
    const float* p = pool + (size_t)g * (2 * HID);
    for (int k = 0; k < 2 * HID; ++k) acc += p[k] * inv * fcW[k * NC + c];
    out[t] = acc;
}

extern "C" void kernel_launch(void* const* d_in, const int* in_sizes, int n_in,
                              void* d_out, int out_size, void* d_ws, size_t ws_size,
                              hipStream_t stream) {
    const float* x       = (const float*)d_in[0];
    const int*   eidx    = (const int*)  d_in[1];   // [2,E] flat: row0=src, row1=dst
    const int*   batch   = (const int*)  d_in[2];
    const int*   rootidx = (const int*)  d_in[3];
    const float* W1      = (const float*)d_in[4];
    const float* b1      = (const float*)d_in[5];
    const float* W2      = (const float*)d_in[6];
    const float* b2      = (const float*)d_in[7];
    const float* fcW     = (const float*)d_in[8];
    const float* fcb     = (const float*)d_in[9];

    const int N  = in_sizes[2];          // 100000
    const int E  = in_sizes[1] / 2;      // 1000000
    const int NG = in_sizes[3];          // 512
    const int NC = in_sizes[9];          // 4

    float* out = (float*)d_out;                    // [NG, NC]
    float* Xf  = out + (size_t)NG * NC;            // X_feat [N, 256]

    // workspace carve-up (~104 MB)
    char* w = (char*)d_ws;
    float* deg   = (float*)w;  w += (size_t)N * 4;            // deg -> dinv (in place)
    int*   rootn = (int*)  w;  w += (size_t)N * 4;
    float* hbuf  = (float*)w;  w += (size_t)N * HID * 4;      // h1, reused as h2
    float* nr1   = (float*)w;  w += (size_t)N * HID * 4;      // node_rep1
    float* pool  = (float*)w;  w += (size_t)NG * 2 * HID * 4; // segment sums
    float* cnt   = (float*)w;                                  // NG counts (contiguous after pool)

    const int B = 256;
    const int gN   = (N + B - 1) / B;
    const int gE   = (E + B - 1) / B;
    const int gNH  = (N * HID + B - 1) / B;
    const int gE32 = (E * 32 + B - 1) / B;

    // degrees + root gather index
    k_prep<<<gN, B, 0, stream>>>(batch, rootidx, deg, rootn, N);
    k_deg <<<gE, B, 0, stream>>>(eidx, deg, E);
    k_dinv<<<gN, B, 0, stream>>>(deg, N);

    // conv1: h1 = x @ W1 ; nr1 = bias + selfloop + edge scatter
    dim3 gb(32, 4);
    dim3 gg((N + 15) / 16, (HID / 16 + 3) / 4);   // (6250, 2)
    k_gemm_wmma<0><<<gg, gb, 0, stream>>>(x, nullptr, nullptr, W1, hbuf, N, HID);
    k_selfbias<<<gNH, B, 0, stream>>>(hbuf, deg, b1, nr1, N, HID);
    k_scatter <<<gE32, B, 0, stream>>>(eidx, hbuf, deg, nr1, E, HID);

    // conv2: h2 = relu(concat(nr1, x[root])) @ W2, scattered straight into X_feat[:, :128]
    k_gemm_wmma<1><<<gg, gb, 0, stream>>>(nr1, x, rootn, W2, hbuf, N, 2 * HID);
    k_selfbias<<<gNH, B, 0, stream>>>(hbuf, deg, b2, Xf, N, 2 * HID);
    k_scatter <<<gE32, B, 0, stream>>>(eidx, hbuf, deg, Xf, E, 2 * HID);

    // relu(g) in place + X_feat[:, 128:] = nr1[root]
    k_relu_fill<<<gNH, B, 0, stream>>>(Xf, nr1, rootn, N);

    // mean pool + FC
    const int poolElems = NG * 2 * HID + NG;
    k_zero<<<(poolElems + B - 1) / B, B, 0, stream>>>(pool, poolElems);
    k_pool<<<(N * 64 + B - 1) / B, B, 0, stream>>>(Xf, batch, pool, cnt, N);
    k_fc  <<<(NG * NC + B - 1) / B, B, 0, stream>>>(pool, cnt, fcW, fcb, out, NG, NC);
}